// ToeplitzConv2d_63539746177447
// MI455X (gfx1250) — compile-verified
//
#include <hip/hip_runtime.h>

typedef __attribute__((ext_vector_type(2))) float v2f;
typedef __attribute__((ext_vector_type(8))) float v8f;

#define IC_    4
#define OC_    8
#define PH_    30
#define PW_    30
#define OH_    28
#define OW_    28
#define XROW   (IC_ * PH_ * PW_)        // 3600
#define OROW   (OC_ * OH_ * OW_)        // 6272
#define OHW    (OH_ * OW_)              // 784
#define BATCH_ 4096
#define XTOTAL (BATCH_ * XROW)          // 14,745,600 (< 2^31)
#define MAXCOL (3 * (PH_ * PW_) + 2 * PW_ + 2)   // 2762 : largest tap offset
#define SAFEIX ((unsigned)(XTOTAL - 1 - MAXCOL)) // clamp target; no-op for store-valid lanes

#define NBLOCKS  1792
#define NWAVES   (NBLOCKS * 8)          // 14336 = 256 * (OH_*2)
#define BSTRIDE  256                    // batch step per wave iteration
#define ITERS    (BATCH_ / BSTRIDE)     // 16

// D = A(16x4 kernel, rows=oc) x B(4x16 x-patch, cols=ow) + C(bias), 9 K-steps (K=36)
// Wave tiling: wave -> fixed (owt, oh, b0); iterates b = b0 + 256*i, i in [0,16).
__global__ __launch_bounds__(256) void toeplitz_conv_wmma(
    const float* __restrict__ x,      // [BATCH, XROW]
    const float* __restrict__ w,      // [OROW, XROW] (Toeplitz; kernel at row oc*784)
    const float* __restrict__ bias,   // [OROW] = per-oc value repeated 784x
    float* __restrict__ out)          // [BATCH, OROW]
{
    const int lane = threadIdx.x & 31;
    const int waveId = blockIdx.x * 8 + (threadIdx.x >> 5);

    const int n   = lane & 15;          // matrix column (ow offset) / A row (oc)
    const int hi  = lane >> 4;          // lane group: K-pair {0,1} vs {2,3}
    const int off = hi << 1;

    // ---- one-time tile coordinates (constant across the wave's 16 iterations) ----
    const int owt = waveId & 1;
    const int oh  = (waveId >> 1) % OH_;
    const int b0  = waveId / (2 * OH_);             // 0..255
    const int ow0 = owt << 4;
    const bool doStore = (hi == 0) & (n < (OW_ - ow0));

    // ---- per-wave constants: A fragments (conv kernel) + tap column offsets ----
    v2f afrag[9];
    int col0[9], col1[9];               // icol(k) = ic*900 + kh*30 + kw  (shared by A and B)
    {
        const int wrow = (n & 7) * OHW;                 // weight row oc*784 (clamped)
        const float as = (n < OC_) ? 1.0f : 0.0f;       // zero-pad oc rows 8..15
#pragma unroll
        for (int s = 0; s < 9; ++s) {
            const int k0 = 4 * s + off;
            const int k1 = k0 + 1;
            const int ic0 = k0 / 9, r0 = k0 - ic0 * 9, kh0 = r0 / 3, kw0 = r0 - kh0 * 3;
            const int ic1 = k1 / 9, r1 = k1 - ic1 * 9, kh1 = r1 / 3, kw1 = r1 - kh1 * 3;
            col0[s] = ic0 * (PH_ * PW_) + kh0 * PW_ + kw0;
            col1[s] = ic1 * (PH_ * PW_) + kh1 * PW_ + kw1;
            afrag[s].x = as * w[wrow * XROW + col0[s]];
            afrag[s].y = as * w[wrow * XROW + col1[s]];
        }
    }

    // C-fragment init values: lanes 0-15 row r = oc r -> bias; lanes 16-31 rows 8..15 -> 0
    float cinit[8];
#pragma unroll
    for (int r = 0; r < 8; ++r) cinit[r] = hi ? 0.0f : bias[r * OHW];

    // running per-lane x index and output pointer (advance by 256 batches per iter)
    unsigned bidx = (unsigned)b0 * XROW + (unsigned)(oh * PW_ + ow0 + n);
    float* __restrict__ orow = out + (size_t)b0 * OROW + (oh * OW_ + ow0 + n);

#pragma unroll 1
    for (int i = 0; i < ITERS; ++i) {
        // single clamp keeps every tap read in-bounds; exact no-op for store-valid lanes
        const unsigned ci = (bidx > SAFEIX) ? SAFEIX : bidx;
        const float* __restrict__ p = x + ci;

        if (i + 1 < ITERS)
            __builtin_prefetch(p + BSTRIDE * XROW, 0, 1);   // next iteration's patch

        v8f acc;
#pragma unroll
        for (int r = 0; r < 8; ++r) acc[r] = cinit[r];

#pragma unroll
        for (int s = 0; s < 9; ++s) {
            v2f bfrag;
            bfrag.x = p[col0[s]];
            bfrag.y = p[col1[s]];
            acc = __builtin_amdgcn_wmma_f32_16x16x4_f32(
                /*neg_a=*/false, afrag[s], /*neg_b=*/false, bfrag,
                /*c_mod=*/(short)0, acc, /*reuse_a=*/false, /*reuse_b=*/false);
        }

        // store: lanes 0-15, D VGPR r = oc r, col = ow0+n ; 16 consecutive floats per oc
        if (doStore) {
#pragma unroll
            for (int r = 0; r < 8; ++r)
                orow[r * OHW] = acc[r];
        }

        bidx += BSTRIDE * XROW;         // next batch group
        orow += (size_t)BSTRIDE * OROW;
    }
}

extern "C" void kernel_launch(void* const* d_in, const int* in_sizes, int n_in,
                              void* d_out, int out_size, void* d_ws, size_t ws_size,
                              hipStream_t stream) {
    const float* x    = (const float*)d_in[0];   // padded_x [4096, 3600]
    const float* w    = (const float*)d_in[1];   // weight   [6272, 3600]
    const float* bias = (const float*)d_in[2];   // bias     [6272]
    float* out = (float*)d_out;                  // [4096, 6272]

    // 1792 blocks x 8 waves = 14336 waves = 256 * 56; each wave owns one
    // (oh, ow-tile) strip and walks 16 batch groups of 256.
    toeplitz_conv_wmma<<<NBLOCKS, 256, 0, stream>>>(x, w, bias, out);
}